// SHNMN_23768349016471
// MI455X (gfx1250) — compile-verified
//
#include <hip/hip_runtime.h>

#define DIMC 64
#define NTOK 3
#define BATCH 256
#define NPIX 256                       // 16*16
#define CNN_W_SZ (DIMC*DIMC*3*3)       // 36864
#define PROJ_W_SZ (2*DIMC*DIMC)        // 8192
#define EMB_DIM (CNN_W_SZ + DIMC + PROJ_W_SZ + DIMC)  // 45184
#define CNN_B_OFF CNN_W_SZ
#define PROJ_W_OFF (CNN_W_SZ + DIMC)
#define PROJ_B_OFF (CNN_W_SZ + DIMC + PROJ_W_SZ)

typedef float v2f __attribute__((ext_vector_type(2)));
typedef float v8f __attribute__((ext_vector_type(8)));

__device__ __forceinline__ v8f wmma_f32(v2f a, v2f b, v8f c) {
    // D = A(16x4,f32) * B(4x16,f32) + C(16x16,f32)
    return __builtin_amdgcn_wmma_f32_16x16x4_f32(false, a, false, b, (short)0, c, false, false);
}

// thread-0 helper: softmax of `len` values from row base, write `outn` slots (zero-pad)
__device__ __forceinline__ void softmax_row(const float* src, int len, float* dst, int outn) {
    float mx = -1e30f;
    for (int t = 0; t < len; ++t) mx = fmaxf(mx, src[t]);
    float s = 0.f;
    for (int t = 0; t < len; ++t) { float e = __expf(src[t] - mx); dst[t] = e; s += e; }
    float inv = 1.f / s;
    for (int t = 0; t < outn; ++t) dst[t] = (t < len) ? dst[t] * inv : 0.f;
}

// ---------------------------------------------------------------------------
// Projection: cnn_inp[b] (64x256) = proj_w[b] (64x128) @ inp[b] (128x256) + proj_b[b]
// inp channels 0..63 = t0-blend of states, 64..127 = t1-blend. Weights blended
// on the fly from 3 gathered q_embed rows (L2 resident).
// ---------------------------------------------------------------------------
__global__ __launch_bounds__(256) void shnmn_proj_kernel(
    int mi,
    const int*   __restrict__ question,
    const float* __restrict__ q_embed,
    const float* __restrict__ alpha,
    const float* __restrict__ tau0,
    const float* __restrict__ tau1,
    const float* __restrict__ src0,    // h_1 = img
    const float* __restrict__ src1,    // h_2 (mi>=1)
    const float* __restrict__ src2,    // h_3 (mi>=2)
    float*       __restrict__ cnn_inp) // (B,64,256)
{
    __shared__ float sA[DIMC * 129];   // blended proj_w, padded stride
    __shared__ float sB[16 * NPIX];    // inp K-panel
    __shared__ float sBias[DIMC];
    __shared__ float sScal[12];        // a[0..2], t0w[4] @4, t1w[4] @8
    __shared__ int   sQ[NTOK];

    const int b   = blockIdx.x;
    const int tid = threadIdx.x;

    if (tid == 0) {
        softmax_row(alpha + mi * 3, 3, sScal, 3);
        softmax_row(tau0  + mi * 4, mi + 2, sScal + 4, 4);
        softmax_row(tau1  + mi * 4, mi + 2, sScal + 8, 4);
        sQ[0] = question[b * NTOK + 0];
        sQ[1] = question[b * NTOK + 1];
        sQ[2] = question[b * NTOK + 2];
    }
    __syncthreads();

    const float a0 = sScal[0], a1 = sScal[1], a2 = sScal[2];
    const float* e0 = q_embed + sQ[0] * EMB_DIM;
    const float* e1 = q_embed + sQ[1] * EMB_DIM;
    const float* e2 = q_embed + sQ[2] * EMB_DIM;

    // blend proj_w into LDS (8192 elems)
    for (int e = tid; e < DIMC * 128; e += 256) {
        int o = e >> 7, c = e & 127;
        int off = PROJ_W_OFF + e;
        sA[o * 129 + c] = a0 * e0[off] + a1 * e1[off] + a2 * e2[off];
    }
    if (tid < DIMC) {
        int off = PROJ_B_OFF + tid;
        sBias[tid] = a0 * e0[off] + a1 * e1[off] + a2 * e2[off];
    }

    const int lane   = tid & 31;
    const int wv     = tid >> 5;       // 0..7
    const int idx    = lane & 15;
    const int kb     = (lane >> 4) * 2;
    const int ntbase = wv * 32;        // 32 output columns per wave

    v8f zero = {0.f, 0.f, 0.f, 0.f, 0.f, 0.f, 0.f, 0.f};
    v8f acc[4][2];
    for (int mt = 0; mt < 4; ++mt)
        for (int nt = 0; nt < 2; ++nt) acc[mt][nt] = zero;

    for (int kp = 0; kp < 8; ++kp) {   // 8 panels of K=16
        __syncthreads();
        // blend inp panel: channels [kp*16, kp*16+16)
        for (int e = tid; e < 16 * NPIX; e += 256) {
            int cl = e >> 8, p = e & 255;
            int c  = kp * 16 + cl;
            const float* w = (c >= DIMC) ? (sScal + 8) : (sScal + 4);
            int cc = c & (DIMC - 1);
            int gidx = (b * DIMC + cc) * NPIX + p;
            float v = w[1] * src0[gidx];
            if (mi >= 1) v += w[2] * src1[gidx];
            if (mi >= 2) v += w[3] * src2[gidx];
            sB[cl * NPIX + p] = v;
        }
        __syncthreads();

        for (int k4 = 0; k4 < 4; ++k4) {
            const int kA = kp * 16 + k4 * 4 + kb;  // column in sA
            const int kB = k4 * 4 + kb;            // row in sB panel
            v2f af[4];
            for (int mt = 0; mt < 4; ++mt) {
                af[mt].x = sA[(mt * 16 + idx) * 129 + kA];
                af[mt].y = sA[(mt * 16 + idx) * 129 + kA + 1];
            }
            for (int nt = 0; nt < 2; ++nt) {
                int n = ntbase + nt * 16 + idx;
                v2f bf;
                bf.x = sB[kB * NPIX + n];
                bf.y = sB[(kB + 1) * NPIX + n];
                for (int mt = 0; mt < 4; ++mt)
                    acc[mt][nt] = wmma_f32(af[mt], bf, acc[mt][nt]);
            }
        }
    }

    for (int mt = 0; mt < 4; ++mt)
        for (int nt = 0; nt < 2; ++nt) {
            int n = ntbase + nt * 16 + idx;
            for (int r = 0; r < 8; ++r) {
                int m = mt * 16 + r + ((lane >> 4) * 8);
                cnn_inp[(b * DIMC + m) * NPIX + n] = acc[mt][nt][r] + sBias[m];
            }
        }
}

// ---------------------------------------------------------------------------
// Conv 3x3 (per-example weights) as implicit GEMM 64x256x(9*64) + bias + relu.
// Whole 64x256 input image staged in LDS, reused by all 9 taps; one 64x64
// blended weight panel per tap.
// ---------------------------------------------------------------------------
__global__ __launch_bounds__(256) void shnmn_conv_kernel(
    int mi,
    const int*   __restrict__ question,
    const float* __restrict__ q_embed,
    const float* __restrict__ alpha,
    const float* __restrict__ cnn_inp, // (B,64,256)
    float*       __restrict__ hout)    // (B,64,256)
{
    __shared__ float sX[DIMC * NPIX];  // 64 KB input image
    __shared__ float sA[DIMC * 65];    // blended tap weights, padded stride
    __shared__ float sBias[DIMC];
    __shared__ float sScal[4];
    __shared__ int   sQ[NTOK];

    const int b   = blockIdx.x;
    const int tid = threadIdx.x;

    if (tid == 0) {
        softmax_row(alpha + mi * 3, 3, sScal, 3);
        sQ[0] = question[b * NTOK + 0];
        sQ[1] = question[b * NTOK + 1];
        sQ[2] = question[b * NTOK + 2];
    }
    __syncthreads();

    const float a0 = sScal[0], a1 = sScal[1], a2 = sScal[2];
    const float* e0 = q_embed + sQ[0] * EMB_DIM;
    const float* e1 = q_embed + sQ[1] * EMB_DIM;
    const float* e2 = q_embed + sQ[2] * EMB_DIM;

    for (int e = tid; e < DIMC * NPIX; e += 256)
        sX[e] = cnn_inp[b * DIMC * NPIX + e];
    if (tid < DIMC) {
        int off = CNN_B_OFF + tid;
        sBias[tid] = a0 * e0[off] + a1 * e1[off] + a2 * e2[off];
    }

    const int lane   = tid & 31;
    const int wv     = tid >> 5;
    const int idx    = lane & 15;
    const int kb     = (lane >> 4) * 2;
    const int ntbase = wv * 32;

    v8f zero = {0.f, 0.f, 0.f, 0.f, 0.f, 0.f, 0.f, 0.f};
    v8f acc[4][2];
    for (int mt = 0; mt < 4; ++mt)
        for (int nt = 0; nt < 2; ++nt) acc[mt][nt] = zero;

    for (int t = 0; t < 9; ++t) {
        const int dy = t / 3 - 1, dx = t % 3 - 1;
        __syncthreads();
        // blend weight panel for this tap: A[o][c] = blend E[(o*64+c)*9 + t]
        for (int e = tid; e < DIMC * DIMC; e += 256) {
            int o = e >> 6, c = e & 63;
            int off = (o * DIMC + c) * 9 + t;
            sA[o * 65 + c] = a0 * e0[off] + a1 * e1[off] + a2 * e2[off];
        }
        __syncthreads();

        int soff[2];
        for (int nt = 0; nt < 2; ++nt) {
            int n = ntbase + nt * 16 + idx;
            int y = (n >> 4) + dy, x = (n & 15) + dx;
            soff[nt] = (y >= 0 && y < 16 && x >= 0 && x < 16) ? (y * 16 + x) : -1;
        }

        for (int k4 = 0; k4 < 16; ++k4) {
            const int k = k4 * 4 + kb;  // input channel
            v2f af[4];
            for (int mt = 0; mt < 4; ++mt) {
                af[mt].x = sA[(mt * 16 + idx) * 65 + k];
                af[mt].y = sA[(mt * 16 + idx) * 65 + k + 1];
            }
            for (int nt = 0; nt < 2; ++nt) {
                v2f bf;
                bf.x = (soff[nt] >= 0) ? sX[k * NPIX + soff[nt]] : 0.f;
                bf.y = (soff[nt] >= 0) ? sX[(k + 1) * NPIX + soff[nt]] : 0.f;
                for (int mt = 0; mt < 4; ++mt)
                    acc[mt][nt] = wmma_f32(af[mt], bf, acc[mt][nt]);
            }
        }
    }

    for (int mt = 0; mt < 4; ++mt)
        for (int nt = 0; nt < 2; ++nt) {
            int n = ntbase + nt * 16 + idx;
            for (int r = 0; r < 8; ++r) {
                int m = mt * 16 + r + ((lane >> 4) * 8);
                hout[(b * DIMC + m) * NPIX + n] = fmaxf(acc[mt][nt][r] + sBias[m], 0.f);
            }
        }
}

extern "C" void kernel_launch(void* const* d_in, const int* in_sizes, int n_in,
                              void* d_out, int out_size, void* d_ws, size_t ws_size,
                              hipStream_t stream) {
    const int*   question = (const int*)  d_in[0];
    const float* img      = (const float*)d_in[1];
    const float* q_embed  = (const float*)d_in[2];
    const float* alpha    = (const float*)d_in[3];
    const float* tau0     = (const float*)d_in[4];
    const float* tau1     = (const float*)d_in[5];
    float* out = (float*)d_out;

    float* wsf     = (float*)d_ws;
    const int HSZ  = BATCH * DIMC * NPIX;   // 4,194,304 floats
    float* cnn_inp = wsf;
    float* h2      = wsf + HSZ;
    float* h3      = wsf + 2 * HSZ;

    dim3 grid(BATCH), blk(256);

    // module 0: lhs/rhs from {sentinel, img}
    shnmn_proj_kernel<<<grid, blk, 0, stream>>>(0, question, q_embed, alpha, tau0, tau1,
                                                img, nullptr, nullptr, cnn_inp);
    shnmn_conv_kernel<<<grid, blk, 0, stream>>>(0, question, q_embed, alpha, cnn_inp, h2);
    // module 1: {sentinel, img, h2}
    shnmn_proj_kernel<<<grid, blk, 0, stream>>>(1, question, q_embed, alpha, tau0, tau1,
                                                img, h2, nullptr, cnn_inp);
    shnmn_conv_kernel<<<grid, blk, 0, stream>>>(1, question, q_embed, alpha, cnn_inp, h3);
    // module 2: {sentinel, img, h2, h3} -> final output
    shnmn_proj_kernel<<<grid, blk, 0, stream>>>(2, question, q_embed, alpha, tau0, tau1,
                                                img, h2, h3, cnn_inp);
    shnmn_conv_kernel<<<grid, blk, 0, stream>>>(2, question, q_embed, alpha, cnn_inp, out);
}